// RNN_30846455120350
// MI455X (gfx1250) — compile-verified
//
#include <hip/hip_runtime.h>
#include <hip/hip_bf16.h>

#define VOCAB  32000
#define HIDDEN 1024
#define BATCH  32
#define SEQ    128

typedef __attribute__((ext_vector_type(2))) float v2f;
typedef __attribute__((ext_vector_type(8))) float v8f;

// ---------------------------------------------------------------------------
// Kernel A: one recurrence step.
//   h_t = tanh(W_xh[x[:,t]] + h_{t-1} @ W_hh + b_h)  -> written into H rows t*32+b
// grid: 16 blocks (64 hidden cols each) x 256 threads (8 waves).
// Per block: 2 M-tiles x 4 N-tiles of 16x16, one tile per wave, K=1024 via
// fp32 WMMA 16x16x4. Two independent accumulator chains per wave (even/odd
// k-quads) so the latency-critical serial step isn't one 256-deep WMMA RAW
// chain at the low occupancy (~2 waves/SIMD) this kernel runs at.
// ---------------------------------------------------------------------------
__global__ __launch_bounds__(256) void rnn_step_kernel(
    const float* __restrict__ h_prev,   // (32,1024)
    const float* __restrict__ W_xh,     // (32000,1024)
    const float* __restrict__ W_hh,     // (1024,1024)
    const float* __restrict__ b_h,      // (1024)
    const int*   __restrict__ x,        // (32,128)
    int t,
    float* __restrict__ H)              // (4096,1024)
{
    __shared__ float hs[BATCH][68];     // 64-wide K chunk, stride 272B: 8B-aligned
                                        // pairs, 16 fragment rows -> 16 banks

    const int tid   = threadIdx.x;
    const int wave  = tid >> 5;
    const int lane  = tid & 31;
    const int l16   = lane & 15;
    const int lhi   = lane >> 4;
    const int mtile = wave >> 2;                    // 0..1  (batch rows 0-15 / 16-31)
    const int ntile = wave & 3;                     // 0..3
    const int ncol0 = blockIdx.x * 64 + ntile * 16; // hidden column base of this tile

    v8f acc0 = {0.f, 0.f, 0.f, 0.f, 0.f, 0.f, 0.f, 0.f};
    v8f acc1 = acc0;

    for (int kc = 0; kc < HIDDEN; kc += 64) {
        __syncthreads();
        for (int i = tid; i < BATCH * 64; i += 256) {
            const int b  = i >> 6;
            const int kk = i & 63;
            hs[b][kk] = h_prev[b * HIDDEN + kc + kk];
        }
        __syncthreads();
#pragma unroll
        for (int ks = 0; ks < 64; ks += 8) {        // two independent quads / iter
            const int kb0 = ks + 2 * lhi;           // chain 0: k-quad at ks
            const int kb1 = ks + 4 + 2 * lhi;       // chain 1: k-quad at ks+4
            v2f a0, a1, b0, b1;
            a0.x = hs[mtile * 16 + l16][kb0];
            a0.y = hs[mtile * 16 + l16][kb0 + 1];
            a1.x = hs[mtile * 16 + l16][kb1];
            a1.y = hs[mtile * 16 + l16][kb1 + 1];
            const float* bp0 = W_hh + (size_t)(kc + kb0) * HIDDEN + ncol0 + l16;
            const float* bp1 = W_hh + (size_t)(kc + kb1) * HIDDEN + ncol0 + l16;
            b0.x = bp0[0];
            b0.y = bp0[HIDDEN];
            b1.x = bp1[0];
            b1.y = bp1[HIDDEN];
            acc0 = __builtin_amdgcn_wmma_f32_16x16x4_f32(
                false, a0, false, b0, (short)0, acc0, false, false);
            acc1 = __builtin_amdgcn_wmma_f32_16x16x4_f32(
                false, a1, false, b1, (short)0, acc1, false, false);
        }
    }

    // Fused embedding gather + bias + tanh, store h_t into H.
    const int   col  = ncol0 + l16;
    const float bias = b_h[col];
#pragma unroll
    for (int r = 0; r < 8; ++r) {
        const int   b   = mtile * 16 + 8 * lhi + r;       // batch row (C/D layout)
        const int   tok = x[b * SEQ + t];
        const float v   = tanhf(acc0[r] + acc1[r] +
                                W_xh[(size_t)tok * HIDDEN + col] + bias);
        H[((size_t)t * BATCH + b) * HIDDEN + col] = v;
    }
}

// ---------------------------------------------------------------------------
// Kernel B: out = H(4096x1024) @ W_ho(1024x32000) + b_o  (memory-bound; fp32 WMMA)
// Block tile 64x128, K-chunk 16, 256 threads = 8 waves, each wave a 32x32 patch
// (2x2 WMMA tiles -> 4 independent accumulator chains). grid(250, 64) covers
// 32000 x 4096 exactly (no divergence, EXEC all-ones for WMMA).
// ---------------------------------------------------------------------------
__global__ __launch_bounds__(256) void out_gemm_kernel(
    const float* __restrict__ H,      // (4096,1024)
    const float* __restrict__ W_ho,   // (1024,32000)
    const float* __restrict__ b_o,    // (32000)
    float* __restrict__ out)          // (4096,32000)
{
    __shared__ float As[64][18];      // stride 72B: 8B-aligned k-pairs (b64 DS
                                      // loads) and r*18 mod 64 injective -> no
                                      // bank conflicts on the 16-row gather
    __shared__ float Bs[16][128];

    const int tid  = threadIdx.x;
    const int wave = tid >> 5;
    const int lane = tid & 31;
    const int l16  = lane & 15;
    const int lhi  = lane >> 4;
    const int wm   = wave >> 2;       // 0..1 -> rows wm*32..+31
    const int wn   = wave & 3;        // 0..3 -> cols wn*32..+31

    const int Mbase = blockIdx.y * 64;
    const int Nbase = blockIdx.x * 128;

    v8f acc00 = {0.f,0.f,0.f,0.f,0.f,0.f,0.f,0.f};
    v8f acc01 = acc00, acc10 = acc00, acc11 = acc00;

    for (int kc = 0; kc < HIDDEN; kc += 16) {
        __syncthreads();
        for (int i = tid; i < 64 * 16; i += 256) {        // 4 per thread, coalesced
            const int row = i >> 4, kk = i & 15;
            As[row][kk] = H[(size_t)(Mbase + row) * HIDDEN + kc + kk];
        }
        for (int i = tid; i < 16 * 128; i += 256) {       // 8 per thread, coalesced
            const int row = i >> 7, col = i & 127;
            Bs[row][col] = W_ho[(size_t)(kc + row) * VOCAB + Nbase + col];
        }
        __syncthreads();
#pragma unroll
        for (int ks = 0; ks < 16; ks += 4) {
            const int kb = ks + 2 * lhi;
            v2f a0, a1, b0, b1;
            a0.x = As[wm * 32 + l16][kb];        a0.y = As[wm * 32 + l16][kb + 1];
            a1.x = As[wm * 32 + 16 + l16][kb];   a1.y = As[wm * 32 + 16 + l16][kb + 1];
            b0.x = Bs[kb][wn * 32 + l16];        b0.y = Bs[kb + 1][wn * 32 + l16];
            b1.x = Bs[kb][wn * 32 + 16 + l16];   b1.y = Bs[kb + 1][wn * 32 + 16 + l16];
            acc00 = __builtin_amdgcn_wmma_f32_16x16x4_f32(false, a0, false, b0, (short)0, acc00, false, false);
            acc01 = __builtin_amdgcn_wmma_f32_16x16x4_f32(false, a0, false, b1, (short)0, acc01, false, false);
            acc10 = __builtin_amdgcn_wmma_f32_16x16x4_f32(false, a1, false, b0, (short)0, acc10, false, false);
            acc11 = __builtin_amdgcn_wmma_f32_16x16x4_f32(false, a1, false, b1, (short)0, acc11, false, false);
        }
    }

    const int   c0  = Nbase + wn * 32 + l16;
    const int   c1  = c0 + 16;
    const float bo0 = b_o[c0];
    const float bo1 = b_o[c1];
#pragma unroll
    for (int r = 0; r < 8; ++r) {
        const int r0 = Mbase + wm * 32 + 8 * lhi + r;
        const int r1 = r0 + 16;
        out[(size_t)r0 * VOCAB + c0] = acc00[r] + bo0;
        out[(size_t)r0 * VOCAB + c1] = acc01[r] + bo1;
        out[(size_t)r1 * VOCAB + c0] = acc10[r] + bo0;
        out[(size_t)r1 * VOCAB + c1] = acc11[r] + bo1;
    }
}

// ---------------------------------------------------------------------------
extern "C" void kernel_launch(void* const* d_in, const int* in_sizes, int n_in,
                              void* d_out, int out_size, void* d_ws, size_t ws_size,
                              hipStream_t stream) {
    const int*   x     = (const int*)  d_in[0];  // (32,128)
    const float* state = (const float*)d_in[1];  // (32,1024) initial h
    const float* W_xh  = (const float*)d_in[2];  // (32000,1024)
    const float* W_hh  = (const float*)d_in[3];  // (1024,1024)
    const float* b_h   = (const float*)d_in[4];  // (1024)
    const float* W_ho  = (const float*)d_in[5];  // (1024,32000)
    const float* b_o   = (const float*)d_in[6];  // (32000)
    float*       out   = (float*)d_out;          // 4096*32000 outputs + 32*1024 final state
    float*       H     = (float*)d_ws;           // (4096,1024) hidden states, 16 MB

    // 128 dependent recurrence steps (graph nodes); step t consumes h_{t-1}.
    for (int t = 0; t < SEQ; ++t) {
        const float* hp = (t == 0) ? state : (H + (size_t)(t - 1) * BATCH * HIDDEN);
        rnn_step_kernel<<<dim3(16), dim3(256), 0, stream>>>(
            hp, W_xh, W_hh, b_h, x, t, H);
    }

    // Big memory-bound output GEMM.
    out_gemm_kernel<<<dim3(VOCAB / 128, (SEQ * BATCH) / 64), dim3(256), 0, stream>>>(
        H, W_ho, b_o, out);

    // Tuple return: append final_state = h_{127} after the outputs.
    hipMemcpyAsync(out + (size_t)SEQ * BATCH * VOCAB,
                   H + (size_t)(SEQ - 1) * BATCH * HIDDEN,
                   (size_t)BATCH * HIDDEN * sizeof(float),
                   hipMemcpyDeviceToDevice, stream);
}